// PointNet2PatchletsSA_12781822673298
// MI455X (gfx1250) — compile-verified
//
#include <hip/hip_runtime.h>
#include <hip/hip_bf16.h>

typedef float v2f __attribute__((ext_vector_type(2)));
typedef float v8f __attribute__((ext_vector_type(8)));

#define NPTS   4096
#define NPOINT 512
#define NSAMP  32
#define R2     0.04f
#define EPSBN  1e-5f
#define CNT_BN 262144.0f   // 2*512*8*32

// ---------------- workspace layout (float offsets) ----------------
#define OFF_X0   0L               // 2*6*512*256   = 1572864
#define OFF_Y0   1572864L         // 2*64*512*256  = 16777216
#define OFF_Y1   18350080L        // 2*64*512*256  = 16777216
#define OFF_WMAX 35127296L        // 2*128*512*8   = 1048576
#define OFF_WMIN 36175872L        // 1048576
#define OFF_NXYZ 37224448L        // 16*512*3      = 24576
#define OFF_P0   37249024L        // 1024*128
#define OFF_P1   37380096L        // 1024*128
#define OFF_P2   37511168L        // 1024*256
#define OFF_S0   37773312L        // 128
#define OFF_S1   37773440L        // 128
#define OFF_S2   37773568L        // 256

// ---------------- helpers ----------------
__device__ __forceinline__ void atomicMaxF(float* p, float v) {
    if (v >= 0.0f) atomicMax((int*)p, __float_as_int(v));
    else           atomicMin((unsigned int*)p, __float_as_uint(v));
}
__device__ __forceinline__ void atomicMinF(float* p, float v) {
    if (v >= 0.0f) atomicMin((int*)p, __float_as_int(v));
    else           atomicMax((unsigned int*)p, __float_as_uint(v));
}

// ---------------- 1) farthest point sampling: one WG per batch-slice ----------------
__global__ __launch_bounds__(1024)
void fps_kernel(const float* __restrict__ xyz, float* __restrict__ nxyz)
{
    extern __shared__ float smem[];
    float* xs   = smem;            // 4096
    float* ys   = xs + NPTS;       // 4096
    float* zs   = ys + NPTS;       // 4096
    float* dist = zs + NPTS;       // 4096
    float* rv   = dist + NPTS;     // 1024
    int*   ri   = (int*)(rv + 1024);
    __shared__ int sfar;

    const int tid = threadIdx.x, bt = blockIdx.x;
    const float* px = xyz + (size_t)(bt*3 + 0) * NPTS;
    const float* py = xyz + (size_t)(bt*3 + 1) * NPTS;
    const float* pz = xyz + (size_t)(bt*3 + 2) * NPTS;

    for (int n = tid; n < NPTS; n += 1024) {
        xs[n] = px[n]; ys[n] = py[n]; zs[n] = pz[n]; dist[n] = 1e10f;
    }
    __syncthreads();

    int far = 0;
    for (int it = 0; it < NPOINT; it++) {
        const float cx = xs[far], cy = ys[far], cz = zs[far];
        if (tid == 0) {
            float* o = &nxyz[(size_t)(bt*NPOINT + it) * 3];
            o[0] = cx; o[1] = cy; o[2] = cz;
        }
        float bv = -1.0f; int bi = 0;
        for (int n = tid; n < NPTS; n += 1024) {
            float dx = xs[n]-cx, dy = ys[n]-cy, dz = zs[n]-cz;
            float dd = dx*dx + dy*dy + dz*dz;
            float nd = fminf(dist[n], dd); dist[n] = nd;
            if (nd > bv) { bv = nd; bi = n; }   // ascending scan -> first-max per thread
        }
        rv[tid] = bv; ri[tid] = bi;
        __syncthreads();
        for (int s = 512; s > 0; s >>= 1) {
            if (tid < s) {
                float v2 = rv[tid+s]; int i2 = ri[tid+s];
                if (v2 > rv[tid] || (v2 == rv[tid] && i2 < ri[tid])) { rv[tid] = v2; ri[tid] = i2; }
            }
            __syncthreads();
        }
        if (tid == 0) sfar = ri[0];
        __syncthreads();
        far = sfar;
        __syncthreads();
    }
}

// ---------------- 2) ball query + grouping: one wave32 per query point ----------------
__global__ __launch_bounds__(256)
void group_kernel(const float* __restrict__ xyz, const float* __restrict__ pts,
                  const float* __restrict__ nxyz, float* __restrict__ x0)
{
    __shared__ int slots[8*32];
    __shared__ int cnts[8];
    const int tid = threadIdx.x, lane = tid & 31, wid = tid >> 5;
    const int q  = blockIdx.x * 8 + wid;
    const int bt = q >> 9, p = q & 511;

    const float* px = xyz + (size_t)(bt*3 + 0) * NPTS;
    const float* py = xyz + (size_t)(bt*3 + 1) * NPTS;
    const float* pz = xyz + (size_t)(bt*3 + 2) * NPTS;
    const float cx = nxyz[(size_t)(bt*NPOINT + p)*3 + 0];
    const float cy = nxyz[(size_t)(bt*NPOINT + p)*3 + 1];
    const float cz = nxyz[(size_t)(bt*NPOINT + p)*3 + 2];
    const float s2 = cx*cx + cy*cy + cz*cz;

    int cnt = 0;
    for (int base = 0; base < NPTS && cnt < NSAMP; base += 32) {
        const int nn = base + lane;
        const float x = px[nn], y = py[nn], z = pz[nn];
        const float sqr = s2 + (x*x + y*y + z*z) - 2.0f*(cx*x + cy*y + cz*z);
        const bool in = !(sqr > R2);
        const unsigned mask = (unsigned)__ballot(in);     // wave32: low 32 bits
        if (in) {
            const int slot = cnt + __popc(mask & ((1u << lane) - 1u));
            if (slot < NSAMP) slots[wid*32 + slot] = nn;
        }
        cnt += __popc(mask);
    }
    if (lane == 0) cnts[wid] = cnt < NSAMP ? cnt : NSAMP;
    __syncthreads();

    const int c = cnts[wid];
    const int j = slots[wid*32 + ((lane < c) ? lane : 0)];
    const int b = bt >> 3, t = bt & 7;

    const float gx = px[j] - cx, gy = py[j] - cy, gz = pz[j] - cz;
    const float* qx = pts + (size_t)(bt*3 + 0) * NPTS;
    const float* qy = pts + (size_t)(bt*3 + 1) * NPTS;
    const float* qz = pts + (size_t)(bt*3 + 2) * NPTS;

    const long base0 = (((long)(b*6 + 0) * 512 + p) * 256) + t*32 + lane;
    const long cs = 512L * 256L;
    x0[base0 + 0*cs] = gx;
    x0[base0 + 1*cs] = gy;
    x0[base0 + 2*cs] = gz;
    x0[base0 + 3*cs] = qx[j];
    x0[base0 + 4*cs] = qy[j];
    x0[base0 + 5*cs] = qz[j];
}

// ---------------- 3) implicit-GEMM conv via V_WMMA_F32_16X16X4_F32 ----------------
// one WG per (b,d) slice; padded input slice in LDS; waves own M-tiles of 16 channels.
template<int CIN, int COUT, bool FUSE>
__global__ __launch_bounds__(256)
void conv_kernel(const float* __restrict__ xin, const float* __restrict__ wgt,
                 float* __restrict__ yout, float* __restrict__ partial,
                 float* __restrict__ wmax, float* __restrict__ wmin)
{
    constexpr int K    = CIN * 64;       // Cin*kh*kw
    constexpr int MT   = COUT / 16;      // M tiles
    constexpr int NGRP = 8 / MT;         // waves per M tile
    constexpr int NT   = 16 / NGRP;      // N tiles per wave
    constexpr int XS   = CIN * 11 * 47;  // padded slice (h: -1..9, w: -7..39)

    extern __shared__ float smem[];
    float* xs   = smem;                  // XS
    float* lbuf = xs + XS;               // 256*8 sums + 256*8 sumsqs
    float* smax = lbuf + 4096;           // COUT*8 (FUSE only)
    float* smin = smax + (FUSE ? COUT*8 : 0);

    const int tid   = threadIdx.x;
    const int slice = blockIdx.x;
    const int b = slice >> 9, d = slice & 511;

    for (int i = tid; i < XS; i += 256) xs[i] = 0.0f;
    if (FUSE) {
        for (int i = tid; i < COUT*8; i += 256) { smax[i] = -3.402823466e38f; smin[i] = 3.402823466e38f; }
    }
    __syncthreads();
    for (int i = tid; i < CIN*256; i += 256) {
        const int cin = i >> 8, r = i & 255, h = r >> 5, w = r & 31;
        xs[(cin*11 + h + 1)*47 + w + 7] = xin[(((long)(b*CIN + cin) * 512 + d) * 256) + r];
    }
    __syncthreads();

    const int lane   = tid & 31, wid = tid >> 5;
    const int mtile  = wid % MT;
    const int ntbase = (wid / MT) * NT;
    const int hi     = lane >> 4;       // which half of the wave
    const int n      = lane & 15;       // A-row (M) and B-col (N) owned by this lane
    const float* wrow = wgt + (size_t)(mtile*16 + n) * K + (hi << 1);

    v8f acc[NT];
    #pragma unroll
    for (int t = 0; t < NT; t++)
        #pragma unroll
        for (int j = 0; j < 8; j++) acc[t][j] = 0.0f;

    for (int k0 = 0; k0 < K; k0 += 4) {
        const v2f a = *(const v2f*)(wrow + k0);            // W[m][k0+koff], W[m][k0+koff+1]
        const int cin = k0 >> 6, rem = k0 & 63;
        const int kh = rem >> 4, kw = rem & 15;
        const int base0 = (cin*11 + kh)*47 + kw + (hi << 1) + n;
        #pragma unroll
        for (int t = 0; t < NT; t++) {
            const int ntile = ntbase + t;
            const int idx = base0 + (ntile >> 1)*47 + ((ntile & 1) << 4);
            v2f bf; bf[0] = xs[idx]; bf[1] = xs[idx + 1];
            acc[t] = __builtin_amdgcn_wmma_f32_16x16x4_f32(
                         false, a, false, bf, (short)0, acc[t], false, false);
        }
    }

    // outputs + deterministic BN partial sums
    float lsum[8], lsq[8];
    #pragma unroll
    for (int r = 0; r < 8; r++) { lsum[r] = 0.0f; lsq[r] = 0.0f; }

    #pragma unroll
    for (int t = 0; t < NT; t++) {
        const int ntile = ntbase + t;
        const int h = ntile >> 1, w = ((ntile & 1) << 4) + n;
        #pragma unroll
        for (int r = 0; r < 8; r++) {
            const float v = acc[t][r];
            const int o = mtile*16 + hi*8 + r;
            lsum[r] += v; lsq[r] += v*v;
            if (FUSE) {
                atomicMaxF(&smax[o*8 + h], v);
                atomicMinF(&smin[o*8 + h], v);
            } else {
                yout[(((long)(b*COUT + o) * 512 + d) * 256) + h*32 + w] = v;
            }
        }
    }
    #pragma unroll
    for (int r = 0; r < 8; r++) { lbuf[tid*8 + r] = lsum[r]; lbuf[2048 + tid*8 + r] = lsq[r]; }
    __syncthreads();

    for (int o = tid; o < COUT; o += 256) {          // fixed-order (deterministic) reduce
        const int mt = o >> 4, h8 = (o >> 3) & 1, r = o & 7;
        float s = 0.0f, qq = 0.0f;
        for (int g = 0; g < NGRP; g++) {
            const int wbase = (mt + g*MT)*32 + h8*16;
            for (int l = 0; l < 16; l++) {
                s  += lbuf[(wbase + l)*8 + r];
                qq += lbuf[2048 + (wbase + l)*8 + r];
            }
        }
        partial[(long)slice * (2*COUT) + o]        = s;
        partial[(long)slice * (2*COUT) + COUT + o] = qq;
    }
    if (FUSE) {
        for (int i = tid; i < COUT*8; i += 256) {
            const int o = i >> 3, h = i & 7;
            const long idx = ((long)(b*COUT + o) * 512 + d) * 8 + h;
            wmax[idx] = smax[i]; wmin[idx] = smin[i];
        }
    }
}

// ---------------- 4) deterministic stats reduction over 1024 blocks ----------------
__global__ __launch_bounds__(256)
void stats_reduce_kernel(const float* __restrict__ partial, float* __restrict__ stats, int COUT)
{
    for (int i = threadIdx.x; i < 2*COUT; i += 256) {
        float s = 0.0f;
        for (int w = 0; w < 1024; w++) s += partial[(long)w * 2 * COUT + i];
        stats[i] = s;
    }
}

// ---------------- 5) in-place batchnorm + relu ----------------
__global__ __launch_bounds__(256)
void bnrelu_kernel(float* __restrict__ y, const float* __restrict__ stats,
                   const float* __restrict__ g, const float* __restrict__ be, int COUT)
{
    const long i = (long)blockIdx.x * 256 + threadIdx.x;
    const int o = (int)((i >> 17) % COUT);          // 512*256 = 2^17 per channel-slab
    const float mean = stats[o] * (1.0f / CNT_BN);
    const float var  = stats[COUT + o] * (1.0f / CNT_BN) - mean*mean;
    const float A  = g[o] * rsqrtf(var + EPSBN);
    const float Bc = be[o] - mean * A;
    const float v = y[i] * A + Bc;
    y[i] = v > 0.0f ? v : 0.0f;
}

// ---------------- 6) finalize: BN+relu on fused max/min, write outputs ----------------
__global__ __launch_bounds__(256)
void finalize_kernel(const float* __restrict__ nxyz,
                     const float* __restrict__ wmax, const float* __restrict__ wmin,
                     const float* __restrict__ stats2,
                     const float* __restrict__ g2, const float* __restrict__ be2,
                     float* __restrict__ out)
{
    const int i = blockIdx.x * 256 + threadIdx.x;   // 0 .. 2^20-1
    if (i < 24576) {                                // new_xyz_out (2,8,3,512)
        const int p = i & 511, c = (i >> 9) % 3, bt = i / 1536;
        out[i] = nxyz[(size_t)(bt*NPOINT + p)*3 + c];
    }
    // feat (2,8,128,512): i = ((b*8+t)*128+o)*512+p
    const int p = i & 511, o = (i >> 9) & 127, t = (i >> 16) & 7, b = i >> 19;
    const float mean = stats2[o] * (1.0f / CNT_BN);
    const float var  = stats2[128 + o] * (1.0f / CNT_BN) - mean*mean;
    const float A  = g2[o] * rsqrtf(var + EPSBN);
    const float Bc = be2[o] - mean * A;
    const long mi = ((long)(b*128 + o) * 512 + p) * 8 + t;
    const float m = (A >= 0.0f) ? wmax[mi] : wmin[mi];   // monotone map -> max commutes
    const float v = A * m + Bc;
    out[24576 + i] = v > 0.0f ? v : 0.0f;
}

// ---------------- host launch ----------------
extern "C" void kernel_launch(void* const* d_in, const int* in_sizes, int n_in,
                              void* d_out, int out_size, void* d_ws, size_t ws_size,
                              hipStream_t stream) {
    const float* xyz = (const float*)d_in[0];
    const float* pts = (const float*)d_in[1];
    const float* w0  = (const float*)d_in[2];
    const float* g0  = (const float*)d_in[4];
    const float* be0 = (const float*)d_in[5];
    const float* w1  = (const float*)d_in[6];
    const float* g1  = (const float*)d_in[8];
    const float* be1 = (const float*)d_in[9];
    const float* w2  = (const float*)d_in[10];
    const float* g2  = (const float*)d_in[12];
    const float* be2 = (const float*)d_in[13];

    float* ws   = (float*)d_ws;
    float* x0   = ws + OFF_X0;
    float* y0   = ws + OFF_Y0;
    float* y1   = ws + OFF_Y1;
    float* wmax = ws + OFF_WMAX;
    float* wmin = ws + OFF_WMIN;
    float* nxyz = ws + OFF_NXYZ;
    float* p0   = ws + OFF_P0;
    float* p1   = ws + OFF_P1;
    float* p2   = ws + OFF_P2;
    float* s0   = ws + OFF_S0;
    float* s1   = ws + OFF_S1;
    float* s2   = ws + OFF_S2;

    const size_t fps_smem = (size_t)(4*NPTS + 1024) * sizeof(float) + 1024 * sizeof(int);
    fps_kernel<<<16, 1024, fps_smem, stream>>>(xyz, nxyz);

    group_kernel<<<1024, 256, 0, stream>>>(xyz, pts, nxyz, x0);

    const size_t sm0 = (size_t)(6*11*47  + 4096) * sizeof(float);
    const size_t sm1 = (size_t)(64*11*47 + 4096) * sizeof(float);
    const size_t sm2 = (size_t)(64*11*47 + 4096 + 128*16) * sizeof(float);

    conv_kernel<6, 64, false><<<1024, 256, sm0, stream>>>(x0, w0, y0, p0, nullptr, nullptr);
    stats_reduce_kernel<<<1, 256, 0, stream>>>(p0, s0, 64);
    bnrelu_kernel<<<65536, 256, 0, stream>>>(y0, s0, g0, be0, 64);

    conv_kernel<64, 64, false><<<1024, 256, sm1, stream>>>(y0, w1, y1, p1, nullptr, nullptr);
    stats_reduce_kernel<<<1, 256, 0, stream>>>(p1, s1, 64);
    bnrelu_kernel<<<65536, 256, 0, stream>>>(y1, s1, g1, be1, 64);

    conv_kernel<64, 128, true><<<1024, 256, sm2, stream>>>(y1, w2, nullptr, p2, wmax, wmin);
    stats_reduce_kernel<<<1, 256, 0, stream>>>(p2, s2, 128);

    finalize_kernel<<<4096, 256, 0, stream>>>(nxyz, wmax, wmin, s2, g2, be2, (float*)d_out);
}